// RGCNClassifier_43989055045966
// MI455X (gfx1250) — compile-verified
//
#include <hip/hip_runtime.h>
#include <hip/hip_bf16.h>

// ---------------------------------------------------------------------------
// RGCN classifier for MI455X (gfx1250, wave32).
// Mean-aggregation commutes with the linear transform: segment-sum x[src]
// per (relation,dst) first, then ONE fused WMMA GEMM:
//   out = sum_r mean_r @ W[r] + x @ root + bias.
// Dense GEMMs use V_WMMA_F32_16X16X4_F32 (full fp32 precision).
// rgcn_transform stages its shared A panels (10 relation tiles + x tile,
// 16x128 f32 each) into LDS via the Tensor Data Mover (TENSOR_LOAD_TO_LDS,
// TENSORcnt-tracked), with D# row padding (128 DW + 4 DW) giving a
// bank-conflict-free 132-float LDS row stride for the WMMA fragment reads.
// ---------------------------------------------------------------------------

typedef float v2f __attribute__((ext_vector_type(2)));
typedef float v8f __attribute__((ext_vector_type(8)));
typedef unsigned u32x4 __attribute__((ext_vector_type(4)));
typedef int i32x4 __attribute__((ext_vector_type(4)));
typedef int i32x8 __attribute__((ext_vector_type(8)));

#define NNODES 50000
#define DDIM   128
#define RREL   10
#define NBASES 4
#define NEDGES 800000
#define NCLS   12
#define NSEL   8192
#define LN_EPS 1e-5f

#define LDS_ROW   132                     // 128 floats + 4 pad (bank-conflict-free)
#define PANEL     (16 * LDS_ROW)          // floats per 16x128 staged tile
#define NPANEL    (RREL + 1)              // 10 relations + x
#define LDS_BYTES (NPANEL * PANEL * 4)    // 92,928 B dynamic LDS

#if __has_include(<hip/amd_detail/amd_gfx1250_TDM.h>)
#define TDM_6ARG 1
#endif

__device__ __forceinline__ float wave_sum32(float v) {
#pragma unroll
  for (int m = 16; m >= 1; m >>= 1) v += __shfl_xor(v, m, 32);
  return v;
}

// Issue one TDM descriptor: 16x128 f32 tile, global (row stride 128) -> LDS
// (row stride 132 via pad_interval=6 [128 DW], pad_amount=3 [4 DW]).
__device__ __forceinline__ void tdm_load_tile_16x128(unsigned lds_byte_off,
                                                     const float* gptr) {
#if __has_builtin(__builtin_amdgcn_tensor_load_to_lds)
  unsigned long long ga = (unsigned long long)(uintptr_t)gptr;
  u32x4 g0;
  g0[0] = 1u;                                  // count=1 (valid user D#)
  g0[1] = lds_byte_off;                        // lds_addr
  g0[2] = (unsigned)ga;                        // global_addr[31:0]
  g0[3] = (unsigned)(ga >> 32) | (2u << 30);   // global_addr[56:32] | type=2
  i32x8 g1;
  g1[0] = (int)((2u << 16)      // data_size = 4B
              | (1u << 20)      // pad_enable
              | (6u << 22)      // pad_interval: 128 DWORDs
              | (3u << 25));    // pad_amount: 4 DWORDs
  g1[1] = (int)(128u << 16);    // tensor_dim0 = 128
  g1[2] = (int)(16u << 16);     // tensor_dim1 = 16
  g1[3] = (int)(128u << 16);    // tile_dim0 = 128
  g1[4] = 16;                   // tile_dim1 = 16 (tile_dim2 = 0)
  g1[5] = 128;                  // tensor_dim0_stride = 128
  g1[6] = 0;
  g1[7] = 0;
  i32x4 z4 = {0, 0, 0, 0};
#if defined(TDM_6ARG)
  i32x8 z8 = {0, 0, 0, 0, 0, 0, 0, 0};
  __builtin_amdgcn_tensor_load_to_lds(g0, g1, z4, z4, z8, 0);
#else
  __builtin_amdgcn_tensor_load_to_lds(g0, g1, z4, z4, 0);
#endif
#else
  (void)lds_byte_off; (void)gptr;
#endif
}

// ---------------------------------------------------------------- utilities
__global__ void zero_f32(float* __restrict__ p, int n) {
  int i = blockIdx.x * blockDim.x + threadIdx.x;
  int s = gridDim.x * blockDim.x;
  for (; i < n; i += s) p[i] = 0.0f;
}

__global__ void zero_u32(unsigned* __restrict__ p, int n) {
  int i = blockIdx.x * blockDim.x + threadIdx.x;
  int s = gridDim.x * blockDim.x;
  for (; i < n; i += s) p[i] = 0u;
}

__global__ void copy_f32(const float* __restrict__ src, float* __restrict__ dst,
                         int n) {
  int i = blockIdx.x * blockDim.x + threadIdx.x;
  int s = gridDim.x * blockDim.x;
  for (; i < n; i += s) dst[i] = src[i];
}

// dst[n*K + k] = src[k*N + n]
__global__ void transpose_f32(const float* __restrict__ src,
                              float* __restrict__ dst, int K, int N) {
  int i = blockIdx.x * blockDim.x + threadIdx.x;
  if (i >= K * N) return;
  int k = i / N, n = i - k * N;
  dst[n * K + k] = src[i];
}

// --------------------------------------------------- edge degree histogram
__global__ void count_edges(const int* __restrict__ dst,
                            const int* __restrict__ etype,
                            unsigned* __restrict__ cnt) {
  int e = blockIdx.x * blockDim.x + threadIdx.x;
  if (e < NEDGES) atomicAdd(&cnt[etype[e] * NNODES + dst[e]], 1u);
}

// ---------------- WT[r][o][i] = sum_b comp[r,b] * bases[b][i][o] (transposed)
__global__ void build_WT(const float* __restrict__ comp,
                         const float* __restrict__ bases,
                         float* __restrict__ WT) {
  int idx = blockIdx.x * blockDim.x + threadIdx.x;
  if (idx >= RREL * DDIM * DDIM) return;
  int r = idx / (DDIM * DDIM);
  int oi = idx - r * (DDIM * DDIM);
  int o = oi / DDIM, i = oi - o * DDIM;
  float acc = 0.0f;
#pragma unroll
  for (int b = 0; b < NBASES; ++b)
    acc += comp[r * NBASES + b] * bases[b * DDIM * DDIM + i * DDIM + o];
  WT[idx] = acc;
}

// ------------------------- per-(relation,dst) segment sum of x[src] rows.
__global__ __launch_bounds__(256) void scatter_edges(
    const float* __restrict__ x, const int* __restrict__ src,
    const int* __restrict__ dst, const int* __restrict__ etype,
    float* __restrict__ sums) {
  int wid = (blockIdx.x * blockDim.x + threadIdx.x) >> 5;  // wave-uniform
  int lane = threadIdx.x & 31;
  if (wid >= NEDGES) return;
  int s = src[wid], d = dst[wid], r = etype[wid];
  const float4 v = *(const float4*)(x + s * DDIM + lane * 4);
  float* base = sums + (r * NNODES + d) * DDIM + lane * 4;
  atomicAdd(base + 0, v.x);
  atomicAdd(base + 1, v.y);
  atomicAdd(base + 2, v.z);
  atomicAdd(base + 3, v.w);
}

// --------------------------------------------------------- generic WMMA GEMM
// C[M,128] = (gather? A[gather[m]] : A[m]) @ B + bias, optional ReLU.
// BT row-major [N][K]. One wave per 16x16 tile; partial M tiles clamp the A
// row (garbage only reaches unstored rows) -> zero predication inside loop.
__global__ __launch_bounds__(256) void wmma_gemm_f32(
    const float* __restrict__ A, int lda, const float* __restrict__ BT,
    const float* __restrict__ bias, float* __restrict__ C, int ldc, int M,
    int K, const int* __restrict__ gather, int relu) {
  int wave = (blockIdx.x * blockDim.x + threadIdx.x) >> 5;
  int lane = threadIdx.x & 31;
  int tile_m = wave >> 3;
  int tile_n = wave & 7;
  int m0 = tile_m * 16;
  if (m0 >= M) return;             // wave-uniform: EXEC all-ones at every WMMA
  int hig = lane >> 4;
  int hi2 = hig * 2;
  int rsel = m0 + (lane & 15);
  if (rsel >= M) rsel = M - 1;
  int arow = gather ? gather[rsel] : rsel;
  int col = tile_n * 16 + (lane & 15);
  const float* ap = A + arow * lda + hi2;
  const float* bp = BT + col * K + hi2;
  v8f acc = {};
#pragma unroll 8
  for (int k0 = 0; k0 < K; k0 += 4) {
    float2 av = *(const float2*)(ap + k0);
    float2 bv = *(const float2*)(bp + k0);
    v2f a; a.x = av.x; a.y = av.y;
    v2f b; b.x = bv.x; b.y = bv.y;
    acc = __builtin_amdgcn_wmma_f32_16x16x4_f32(false, a, false, b, (short)0,
                                                acc, false, false);
  }
  float bval = bias ? bias[col] : 0.0f;
#pragma unroll
  for (int v = 0; v < 8; ++v) {
    int row = m0 + v + 8 * hig;
    if (row < M) {
      float o = acc[v] + bval;
      if (relu) o = fmaxf(o, 0.0f);
      C[row * ldc + col] = o;
    }
  }
}

// ------------------------------------------- fused RGCN transform (WMMA+TDM)
// out[n,:] = sum_r (sums[r,n,:]/max(cnt[r,n],1)) @ W[r] + x[n,:] @ root + bias
// Block = 8 waves = one 16-node tile x 8 column tiles. The 11 shared A panels
// are DMA'd into LDS once per block by wave 0 (TENSOR_LOAD_TO_LDS), then each
// wave feeds WMMA from ds_load_b64 (A) + global_load_b64 (B^T, L2-resident).
__global__ __launch_bounds__(256) void rgcn_transform(
    const float* __restrict__ sums, const unsigned* __restrict__ cnt,
    const float* __restrict__ x, const float* __restrict__ WT,
    const float* __restrict__ rootT, const float* __restrict__ bias,
    float* __restrict__ out) {
  extern __shared__ float smem[];  // NPANEL * PANEL floats (dynamic LDS)
  int wave = (blockIdx.x * blockDim.x + threadIdx.x) >> 5;
  int lane = threadIdx.x & 31;
  int tile_m = wave >> 3;
  int tile_n = wave & 7;
  if (tile_m >= NNODES / 16) return;
  int m0 = tile_m * 16;
  int row = m0 + (lane & 15);
  int hig = lane >> 4;
  int hi2 = hig * 2;
  int col = tile_n * 16 + (lane & 15);

#if __has_builtin(__builtin_amdgcn_tensor_load_to_lds)
  if (threadIdx.x < 32) {  // wave 0 drives the Tensor Data Mover
    unsigned lds0 = __builtin_amdgcn_groupstaticsize();  // dynamic-LDS base
#pragma unroll
    for (int r = 0; r < RREL; ++r)
      tdm_load_tile_16x128(lds0 + r * (PANEL * 4),
                           sums + (r * NNODES + m0) * DDIM);
    tdm_load_tile_16x128(lds0 + RREL * (PANEL * 4), x + m0 * DDIM);
#if __has_builtin(__builtin_amdgcn_s_wait_tensorcnt)
    __builtin_amdgcn_s_wait_tensorcnt(0);
#endif
  }
  __syncthreads();
#else
  // Fallback: cooperative staging into the same padded LDS layout.
  for (int idx = threadIdx.x; idx < NPANEL * 16 * 64; idx += 256) {
    int p = idx / (16 * 64);
    int rem = idx - p * (16 * 64);
    int rr = rem >> 6;
    int c2 = rem & 63;
    const float* srcp = (p < RREL) ? sums + (p * NNODES + m0 + rr) * DDIM
                                   : x + (m0 + rr) * DDIM;
    *(float2*)(smem + p * PANEL + rr * LDS_ROW + c2 * 2) =
        *(const float2*)(srcp + c2 * 2);
  }
  __syncthreads();
#endif

  v8f acc = {};
#pragma unroll 1
  for (int r = 0; r < RREL; ++r) {
    float scale = 1.0f / fmaxf((float)cnt[r * NNODES + row], 1.0f);
    const float* ap = smem + r * PANEL + (lane & 15) * LDS_ROW + hi2;
    const float* bp = WT + (r * DDIM + col) * DDIM + hi2;
#pragma unroll 8
    for (int k0 = 0; k0 < DDIM; k0 += 4) {
      float2 av = *(const float2*)(ap + k0);
      float2 bv = *(const float2*)(bp + k0);
      v2f a; a.x = av.x * scale; a.y = av.y * scale;
      v2f b; b.x = bv.x; b.y = bv.y;
      acc = __builtin_amdgcn_wmma_f32_16x16x4_f32(false, a, false, b, (short)0,
                                                  acc, false, false);
    }
  }
  {  // self/root term
    const float* ap = smem + RREL * PANEL + (lane & 15) * LDS_ROW + hi2;
    const float* bp = rootT + col * DDIM + hi2;
#pragma unroll 8
    for (int k0 = 0; k0 < DDIM; k0 += 4) {
      float2 av = *(const float2*)(ap + k0);
      float2 bv = *(const float2*)(bp + k0);
      v2f a; a.x = av.x; a.y = av.y;
      v2f b; b.x = bv.x; b.y = bv.y;
      acc = __builtin_amdgcn_wmma_f32_16x16x4_f32(false, a, false, b, (short)0,
                                                  acc, false, false);
    }
  }
  float bval = bias[col];
#pragma unroll
  for (int v = 0; v < 8; ++v)
    out[(m0 + v + 8 * hig) * DDIM + col] = acc[v] + bval;
}

// ------------------------------------------------ LayerNorm (+optional ReLU)
__global__ __launch_bounds__(256) void layernorm_kernel(
    const float* __restrict__ in, const float* __restrict__ g,
    const float* __restrict__ b, float* __restrict__ out, int n, int relu) {
  int wave = (blockIdx.x * blockDim.x + threadIdx.x) >> 5;
  int lane = threadIdx.x & 31;
  if (wave >= n) return;
  const float4 v = *(const float4*)(in + wave * DDIM + lane * 4);
  float mean = wave_sum32(v.x + v.y + v.z + v.w) * (1.0f / DDIM);
  float dx = v.x - mean, dy = v.y - mean, dz = v.z - mean, dw = v.w - mean;
  float var = wave_sum32(dx * dx + dy * dy + dz * dz + dw * dw) * (1.0f / DDIM);
  float inv = rsqrtf(var + LN_EPS);
  float4 gg = *(const float4*)(g + lane * 4);
  float4 bb = *(const float4*)(b + lane * 4);
  float4 o;
  o.x = dx * inv * gg.x + bb.x;
  o.y = dy * inv * gg.y + bb.y;
  o.z = dz * inv * gg.z + bb.z;
  o.w = dw * inv * gg.w + bb.w;
  if (relu) {
    o.x = fmaxf(o.x, 0.0f); o.y = fmaxf(o.y, 0.0f);
    o.z = fmaxf(o.z, 0.0f); o.w = fmaxf(o.w, 0.0f);
  }
  *(float4*)(out + wave * DDIM + lane * 4) = o;
}

// ----------------------------------------- final 128 -> 12 classifier head
__global__ __launch_bounds__(256) void classifier_out(
    const float* __restrict__ t, const float* __restrict__ Wc2,
    const float* __restrict__ bc2, float* __restrict__ out) {
  int wave = (blockIdx.x * blockDim.x + threadIdx.x) >> 5;
  int lane = threadIdx.x & 31;
  if (wave >= NSEL) return;
  const float4 h = *(const float4*)(t + wave * DDIM + lane * 4);
  int k = lane * 4;
#pragma unroll
  for (int c = 0; c < NCLS; ++c) {
    float p = h.x * Wc2[(k + 0) * NCLS + c] + h.y * Wc2[(k + 1) * NCLS + c] +
              h.z * Wc2[(k + 2) * NCLS + c] + h.w * Wc2[(k + 3) * NCLS + c];
    p = wave_sum32(p);
    if (lane == 0) out[wave * NCLS + c] = p + bc2[c];
  }
}

// ---------------------------------------------------------------------------
static inline int gemm_blocks(int M) { return (M + 15) / 16; }

extern "C" void kernel_launch(void* const* d_in, const int* in_sizes, int n_in,
                              void* d_out, int out_size, void* d_ws,
                              size_t ws_size, hipStream_t stream) {
  const int* edge_index = (const int*)d_in[0];
  const int* e_src = edge_index;
  const int* e_dst = edge_index + NEDGES;
  const int* e_typ = (const int*)d_in[1];
  const int* node_indices = (const int*)d_in[2];
  const float* file_feats = (const float*)d_in[3];
  const float* domain_feats = (const float*)d_in[5];
  const float* ip_feats = (const float*)d_in[7];
  const float* fallback = (const float*)d_in[9];
  const float* Wf = (const float*)d_in[10]; const float* bf = (const float*)d_in[11];
  const float* Wd = (const float*)d_in[12]; const float* bd = (const float*)d_in[13];
  const float* Wi = (const float*)d_in[14]; const float* bi = (const float*)d_in[15];
  const float* comp1 = (const float*)d_in[16]; const float* bases1 = (const float*)d_in[17];
  const float* root1 = (const float*)d_in[18]; const float* bias1 = (const float*)d_in[19];
  const float* comp2 = (const float*)d_in[20]; const float* bases2 = (const float*)d_in[21];
  const float* root2 = (const float*)d_in[22]; const float* bias2 = (const float*)d_in[23];
  const float* ln1_g = (const float*)d_in[24]; const float* ln1_b = (const float*)d_in[25];
  const float* ln2_g = (const float*)d_in[26]; const float* ln2_b = (const float*)d_in[27];
  const float* Wc1 = (const float*)d_in[28]; const float* bc1 = (const float*)d_in[29];
  const float* Wc2 = (const float*)d_in[30]; const float* bc2 = (const float*)d_in[31];
  float* out = (float*)d_out;

  // ---- workspace layout (floats) ----
  const int ND = NNODES * DDIM;                              // 6.4M
  float* x = (float*)d_ws;                                   // [N,D]
  float* h = x + ND;                                         // [N,D]
  float* h2 = h + ND;                                        // [N,D]
  float* sums = h2 + ND;                                     // [R,N,D]
  unsigned* cnt = (unsigned*)(sums + RREL * (size_t)ND);     // [R*N]
  float* WT = (float*)(cnt + RREL * NNODES);                 // [R,D,D] transposed
  float* t = WT + RREL * DDIM * DDIM;                        // [8192,128]
  float* WfT = t + NSEL * DDIM;                              // [128,256]
  float* WdT = WfT + 256 * DDIM;                             // [128,128]
  float* WiT = WdT + 128 * DDIM;                             // [128,64]
  float* rT1 = WiT + 64 * DDIM;                              // [128,128]
  float* rT2 = rT1 + DDIM * DDIM;                            // [128,128]
  float* Wc1T = rT2 + DDIM * DDIM;                           // [128,128]

  // ---- 0. one-shot weight transposes (tiny) ----
  transpose_f32<<<(256 * DDIM + 255) / 256, 256, 0, stream>>>(Wf, WfT, 256, DDIM);
  transpose_f32<<<(128 * DDIM + 255) / 256, 256, 0, stream>>>(Wd, WdT, 128, DDIM);
  transpose_f32<<<(64 * DDIM + 255) / 256, 256, 0, stream>>>(Wi, WiT, 64, DDIM);
  transpose_f32<<<(DDIM * DDIM + 255) / 256, 256, 0, stream>>>(root1, rT1, DDIM, DDIM);
  transpose_f32<<<(DDIM * DDIM + 255) / 256, 256, 0, stream>>>(root2, rT2, DDIM, DDIM);
  transpose_f32<<<(DDIM * DDIM + 255) / 256, 256, 0, stream>>>(Wc1, Wc1T, DDIM, DDIM);

  // ---- 1. node feature construction (typed projections over fallback) ----
  wmma_gemm_f32<<<gemm_blocks(20000), 256, 0, stream>>>(
      file_feats, 256, WfT, bf, x, DDIM, 20000, 256, nullptr, 0);
  wmma_gemm_f32<<<gemm_blocks(15000), 256, 0, stream>>>(
      domain_feats, 128, WdT, bd, x + 20000 * DDIM, DDIM, 15000, 128, nullptr, 0);
  wmma_gemm_f32<<<gemm_blocks(10000), 256, 0, stream>>>(
      ip_feats, 64, WiT, bi, x + 35000 * DDIM, DDIM, 10000, 64, nullptr, 0);
  copy_f32<<<2500, 256, 0, stream>>>(fallback + 45000 * DDIM, x + 45000 * DDIM,
                                     5000 * DDIM);

  // ---- 2. degree counts per (relation, dst) — same for both layers ----
  zero_u32<<<1954, 256, 0, stream>>>(cnt, RREL * NNODES);
  count_edges<<<(NEDGES + 255) / 256, 256, 0, stream>>>(e_dst, e_typ, cnt);

  // ---- 3. layer 1 ----
  zero_f32<<<4096, 256, 0, stream>>>(sums, RREL * ND);
  build_WT<<<(RREL * DDIM * DDIM + 255) / 256, 256, 0, stream>>>(comp1, bases1, WT);
  scatter_edges<<<NEDGES / 8, 256, 0, stream>>>(x, e_src, e_dst, e_typ, sums);
  rgcn_transform<<<NNODES / 16, 256, LDS_BYTES, stream>>>(sums, cnt, x, WT, rT1,
                                                          bias1, h2);
  layernorm_kernel<<<(NNODES + 7) / 8, 256, 0, stream>>>(h2, ln1_g, ln1_b, h,
                                                         NNODES, /*relu=*/1);

  // ---- 4. layer 2 ----
  zero_f32<<<4096, 256, 0, stream>>>(sums, RREL * ND);
  build_WT<<<(RREL * DDIM * DDIM + 255) / 256, 256, 0, stream>>>(comp2, bases2, WT);
  scatter_edges<<<NEDGES / 8, 256, 0, stream>>>(h, e_src, e_dst, e_typ, sums);
  rgcn_transform<<<NNODES / 16, 256, LDS_BYTES, stream>>>(sums, cnt, h, WT, rT2,
                                                          bias2, h2);
  layernorm_kernel<<<(NNODES + 7) / 8, 256, 0, stream>>>(h2, ln2_g, ln2_b, x,
                                                         NNODES, /*relu=*/0);

  // ---- 5. classifier head on gathered rows ----
  wmma_gemm_f32<<<gemm_blocks(NSEL), 256, 0, stream>>>(
      x, DDIM, Wc1T, bc1, t, DDIM, NSEL, DDIM, node_indices, /*relu=*/1);
  classifier_out<<<NSEL / 8, 256, 0, stream>>>(t, Wc2, bc2, out);
}